// ClassCapsule_9466107921106
// MI455X (gfx1250) — compile-verified
//
#include <hip/hip_runtime.h>

typedef __attribute__((ext_vector_type(2))) float v2f;
typedef __attribute__((ext_vector_type(4))) float v4f;
typedef __attribute__((ext_vector_type(8))) float v8f;

#define R_DIM 2048
#define C_DIM 32
#define OUT_DIM 32
#define IN_DIM 16

// ---------------------------------------------------------------------------
// Kernel 1: votes[b,r,c,o] = sum_i vote[r,c,o,i] * x[b,r,i]
// One wave per (r, c). GEMM per (r,c): M=32 (o, 2 tiles), N=16 (b-half),
// K=16 (i, 4 chunks) of V_WMMA_F32_16X16X4_F32.
//   A[m=o, k=i] = vote[r,c,o,i]     B[k=i, n=b] = x[bh*16+b, r, i]
// D layout => lane holds 8 consecutive o for fixed b in 8 consecutive VGPRs,
// so results store as 4x global_store_b128 off a single base address.
// ws layout: [R][bl(16)][cl(cc)][OUT]
// ---------------------------------------------------------------------------
__global__ void votes_kernel(const float* __restrict__ x,
                             const float* __restrict__ vote,
                             float* __restrict__ ws,
                             int c0, int cc, int bh) {
    const int wid  = blockIdx.x * (blockDim.x >> 5) + (threadIdx.x >> 5);
    const int lane = threadIdx.x & 31;
    const int r  = wid / cc;          // wave-uniform
    const int cl = wid - r * cc;      // wave-uniform
    if (r >= R_DIM) return;           // uniform guard: EXEC stays all-1s
    const int c  = c0 + cl;

    const int lm = lane & 15;   // 0..15
    const int lh = lane >> 4;   // 0..1

    // B operand: x[bh*16 + lm, r, i]  (n = b = lm)
    const float* xrow = x + (((size_t)(bh * 16 + lm)) * R_DIM + r) * IN_DIM;
    // A operand: vote[r, c, o, i]     (m = o; tile 0: o=lm, tile 1: o=16+lm)
    const float* vrow = vote + (((size_t)r * C_DIM + c) * OUT_DIM + lm) * IN_DIM;

    v8f acc0 = {};  // o = 0..15  tile
    v8f acc1 = {};  // o = 16..31 tile

#pragma unroll
    for (int kk = 0; kk < 4; ++kk) {
        const int ib = 4 * kk + 2 * lh;            // i base for this lane-half
        v2f bx = *(const v2f*)(xrow + ib);
        v2f a0 = *(const v2f*)(vrow + ib);
        v2f a1 = *(const v2f*)(vrow + 16 * IN_DIM + ib);
        acc0 = __builtin_amdgcn_wmma_f32_16x16x4_f32(false, a0, false, bx,
                                                     (short)0, acc0, false, false);
        acc1 = __builtin_amdgcn_wmma_f32_16x16x4_f32(false, a1, false, bx,
                                                     (short)0, acc1, false, false);
    }

    // Lane (bl = lm) holds o = mt*16 + 8*lh + j  for j = 0..7, mt = 0,1.
    float* dst = ws + (((size_t)r * 16 + lm) * cc + cl) * OUT_DIM + 8 * lh;
    *(v4f*)(dst + 0)  = __builtin_shufflevector(acc0, acc0, 0, 1, 2, 3);
    *(v4f*)(dst + 4)  = __builtin_shufflevector(acc0, acc0, 4, 5, 6, 7);
    *(v4f*)(dst + 16) = __builtin_shufflevector(acc1, acc1, 0, 1, 2, 3);
    *(v4f*)(dst + 20) = __builtin_shufflevector(acc1, acc1, 4, 5, 6, 7);
}

// ---------------------------------------------------------------------------
// Kernel 2: dynamic routing for one (b, c) pair per block (256 threads).
// agreement a[r] and exp-table live in LDS; softmax over R=2048 via tree
// reductions; squash over OUT=32; agreement update fused with float4 reads.
// V row r at: ws + ((r*16 + bl)*cc + cl)*OUT  (stride 16*cc*OUT floats)
// ---------------------------------------------------------------------------
__global__ void routing_kernel(const float* __restrict__ ws,
                               float* __restrict__ out,
                               int c0, int cc, int bh) {
    __shared__ float a_lds[R_DIM];
    __shared__ float e_lds[R_DIM];
    __shared__ float red[256];
    __shared__ float part[256];
    __shared__ float summary[OUT_DIM];
    __shared__ float verdict[OUT_DIM];

    const int t  = threadIdx.x;
    const int bl = blockIdx.x;   // 0..15
    const int cl = blockIdx.y;   // 0..cc-1
    const size_t rstride = (size_t)16 * cc * OUT_DIM;
    const float* V = ws + ((size_t)bl * cc + cl) * OUT_DIM;

    for (int k = 0; k < 8; ++k) a_lds[t + 256 * k] = 0.0f;
    __syncthreads();

    for (int iter = 0; iter < 3; ++iter) {
        // ---- block max of agreement (softmax stabilization) ----
        float m = -1e30f;
        for (int k = 0; k < 8; ++k) m = fmaxf(m, a_lds[t + 256 * k]);
        red[t] = m; __syncthreads();
        for (int off = 128; off > 0; off >>= 1) {
            if (t < off) red[t] = fmaxf(red[t], red[t + off]);
            __syncthreads();
        }
        m = red[0]; __syncthreads();

        // ---- exp table + block sum ----
        float sl = 0.0f;
        for (int k = 0; k < 8; ++k) {
            float e = __expf(a_lds[t + 256 * k] - m);
            e_lds[t + 256 * k] = e;
            sl += e;
        }
        red[t] = sl; __syncthreads();
        for (int off = 128; off > 0; off >>= 1) {
            if (t < off) red[t] += red[t + off];
            __syncthreads();
        }
        const float s = red[0]; __syncthreads();

        // ---- summary[o] = sum_r softmax(a)[r] * V[r,o] ----
        const int g = t >> 5, o = t & 31;       // one wave == one r-stripe
        float p = 0.0f;
        const int rbase = g * 256;
        for (int rr = 0; rr < 256; ++rr) {
            const int r = rbase + rr;
            p += e_lds[r] * V[(size_t)r * rstride + o];   // 128B coalesced/wave
        }
        part[t] = p; __syncthreads();
        if (t < 32) {
            float sm = 0.0f;
            for (int gg = 0; gg < 8; ++gg) sm += part[gg * 32 + t];
            summary[t] = sm / s;
        }
        __syncthreads();

        // ---- squash over OUT ----
        if (t < 32) {
            float sq = 0.0f;
            for (int oo = 0; oo < OUT_DIM; ++oo) sq += summary[oo] * summary[oo];
            const float scale = (sq / (1.0f + sq)) * rsqrtf(sq + 1e-8f);
            verdict[t] = summary[t] * scale;
        }
        __syncthreads();

        // ---- agreement update (last iteration's update is unused) ----
        if (iter < 2) {
            for (int k = 0; k < 8; ++k) {
                const int r = t + 256 * k;
                const float* Vr = V + (size_t)r * rstride;
                float d = 0.0f;
#pragma unroll
                for (int o4 = 0; o4 < 8; ++o4) {
                    v4f w = *(const v4f*)(Vr + 4 * o4);
                    d += w.x * verdict[4 * o4 + 0] + w.y * verdict[4 * o4 + 1] +
                         w.z * verdict[4 * o4 + 2] + w.w * verdict[4 * o4 + 3];
                }
                a_lds[r] += d;
            }
            __syncthreads();
        }
    }

    if (t < 32) {
        const int bg = bh * 16 + bl;
        const int cg = c0 + cl;
        out[((size_t)bg * C_DIM + cg) * OUT_DIM + t] = verdict[t];
    }
}

// ---------------------------------------------------------------------------
// Host: chunk over (b-half, C) so the votes intermediate fits d_ws.
// Per c-slice (16 b's): 16*2048*32*4 = 4 MB of workspace.
// ---------------------------------------------------------------------------
extern "C" void kernel_launch(void* const* d_in, const int* in_sizes, int n_in,
                              void* d_out, int out_size, void* d_ws, size_t ws_size,
                              hipStream_t stream) {
    (void)in_sizes; (void)n_in; (void)out_size;
    const float* x    = (const float*)d_in[0];   // [32, 2048, 16]
    const float* vote = (const float*)d_in[1];   // [2048, 32, 32, 16]
    float* out = (float*)d_out;                  // [32, 32, 32]
    float* ws  = (float*)d_ws;

    const size_t per_c = (size_t)16 * R_DIM * OUT_DIM * sizeof(float);  // 4 MB
    int chunkC = (int)(ws_size / per_c);
    if (chunkC < 1) chunkC = 1;
    if (chunkC > C_DIM) chunkC = C_DIM;

    for (int bh = 0; bh < 2; ++bh) {
        for (int c0 = 0; c0 < C_DIM; c0 += chunkC) {
            const int cc = (c0 + chunkC <= C_DIM) ? chunkC : (C_DIM - c0);
            const int waves  = R_DIM * cc;       // one wave per (r, c)
            const int blocks = waves / 8;        // 256 threads = 8 waves
            hipLaunchKernelGGL(votes_kernel, dim3(blocks), dim3(256), 0, stream,
                               x, vote, ws, c0, cc, bh);
            hipLaunchKernelGGL(routing_kernel, dim3(16, cc), dim3(256), 0, stream,
                               ws, out, c0, cc, bh);
        }
    }
}